// LinearAttention_7584912245416
// MI455X (gfx1250) — compile-verified
//
#include <hip/hip_runtime.h>

typedef float v2f __attribute__((ext_vector_type(2)));
typedef float v8f __attribute__((ext_vector_type(8)));

#define NPIX 4096   // 64*64
#define CDIM 128

// ---------------------------------------------------------------------------
// Generic fp32 WMMA GEMM:  Out[b] (rows mbase..mbase+63) = A[b](64x128) @ X[b](128x4096)
// block = 256 threads (8 waves). Each wave computes two 16x16 C tiles.
// A tile is filled by GLOBAL_LOAD_ASYNC_TO_LDS_B128 (overlapped with B fetch).
// B is stored K-major (lbT[n*68+k]) so the WMMA B operand is one ds_load_b64.
// grid: (4096/64 nslabs, Mchunks, 32 batches)
// ---------------------------------------------------------------------------
__global__ __launch_bounds__(256) void wmma_gemm_kernel(
    const float* __restrict__ Aall, long aBatchStride,
    const float* __restrict__ X,
    float* __restrict__ Out, long outBatchStride,
    const float* __restrict__ bias)
{
    const int b     = blockIdx.z;
    const int mbase = blockIdx.y * 64;
    const int nb0   = blockIdx.x * 64;

    const float* A  = Aall + (size_t)b * aBatchStride + (size_t)mbase * CDIM;
    const float* Xb = X   + (size_t)b * CDIM * NPIX;
    float*       Ob = Out + (size_t)b * outBatchStride + (size_t)mbase * NPIX;

    __shared__ float la [64 * 132];  // [m][k] pitch 132: 16B-aligned chunks, conflict-free b64 reads
    __shared__ float lbT[64 * 68];   // [n][k] pitch 68:  K-major, conflict-free b64 reads

    const int tid = threadIdx.x;

    // ---- async copy A tile (64x128) into la: 16B per lane, pitched dest ----
    #pragma unroll
    for (int it = 0; it < 8; ++it) {
        int i  = tid + it * 256;
        int r  = i >> 5, c4 = i & 31;
        unsigned loff = (unsigned)(size_t)&la[r * 132 + c4 * 4];
        const float* src = &A[r * CDIM + c4 * 4];
        asm volatile("global_load_async_to_lds_b128 %0, %1, off"
                     :: "v"(loff), "v"(src) : "memory");
    }

    const int lane = tid & 31;
    const int wave = tid >> 5;
    const int t0  = wave * 2;          // two adjacent tiles share tm
    const int tm  = t0 >> 2;
    const int tn0 = t0 & 3;
    const int n16 = lane & 15;
    const int klo = (lane >> 4) * 2;   // K-pair select per lane half
    const int arow  = (tm * 16 + n16) * 132;
    const int brow0 = (tn0 * 16 + n16) * 68;
    const int brow1 = brow0 + 16 * 68;

    v8f c0 = {}; v8f c1 = {};

    for (int kh = 0; kh < 2; ++kh) {
        // fetch B K-half (64k x 64n) to registers (overlaps async A on kh==0)
        float4 breg[4];
        #pragma unroll
        for (int it = 0; it < 4; ++it) {
            int i = tid + it * 256;
            int r = i >> 4, n4 = i & 15;
            breg[it] = *(const float4*)&Xb[(size_t)(kh * 64 + r) * NPIX + nb0 + n4 * 4];
        }
        __syncthreads();   // all waves done reading lbT from previous half
        #pragma unroll
        for (int it = 0; it < 4; ++it) {
            int i = tid + it * 256;
            int r = i >> 4, n4 = i & 15;
            lbT[(n4 * 4 + 0) * 68 + r] = breg[it].x;
            lbT[(n4 * 4 + 1) * 68 + r] = breg[it].y;
            lbT[(n4 * 4 + 2) * 68 + r] = breg[it].z;
            lbT[(n4 * 4 + 3) * 68 + r] = breg[it].w;
        }
        if (kh == 0)
            asm volatile("s_wait_asynccnt 0x0" ::: "memory");  // A tile resident (per wave)
        __syncthreads();   // lbT stores + A async visible to all waves

        const int kbase = kh * 64;
        #pragma unroll
        for (int kk = 0; kk < 64; kk += 4) {
            v2f av  = *(const v2f*)&la [arow  + kbase + kk + klo];
            v2f bv0 = *(const v2f*)&lbT[brow0 + kk + klo];
            v2f bv1 = *(const v2f*)&lbT[brow1 + kk + klo];
            c0 = __builtin_amdgcn_wmma_f32_16x16x4_f32(false, av, false, bv0, (short)0, c0, false, false);
            c1 = __builtin_amdgcn_wmma_f32_16x16x4_f32(false, av, false, bv1, (short)0, c1, false, false);
        }
    }

    // store: C/D layout: VGPR i -> row m = i + 8*(lane>=16), col = lane&15
    const int half = lane >> 4;
    #pragma unroll
    for (int i = 0; i < 8; ++i) {
        int m = tm * 16 + i + half * 8;
        float bb = bias ? bias[mbase + m] : 0.0f;
        Ob[(size_t)m * NPIX + nb0 + tn0 * 16 + n16]      = c0[i] + bb;
        Ob[(size_t)m * NPIX + nb0 + tn0 * 16 + 16 + n16] = c1[i] + bb;
    }
}

// ---------------------------------------------------------------------------
// Per-row online softmax stats for k: max and sum(exp(x-max)). 4096 rows.
// ---------------------------------------------------------------------------
__global__ __launch_bounds__(256) void row_stats_kernel(
    const float* __restrict__ kvbuf, float* __restrict__ stats)
{
    const int row = blockIdx.x;                 // b*128 + r
    const int b = row >> 7, r = row & 127;
    const float* kr = kvbuf + ((size_t)b * 256 + r) * NPIX;

    float m = -1e30f, s = 0.0f;
    for (int n = threadIdx.x; n < NPIX; n += 256) {
        float val = kr[n];
        if (val > m) { s = s * __expf(m - val) + 1.0f; m = val; }
        else          { s += __expf(val - m); }
    }
    __shared__ float sm[256], ss[256];
    sm[threadIdx.x] = m; ss[threadIdx.x] = s;
    __syncthreads();
    for (int off = 128; off; off >>= 1) {
        if (threadIdx.x < off) {
            float m1 = sm[threadIdx.x], s1 = ss[threadIdx.x];
            float m2 = sm[threadIdx.x + off], s2 = ss[threadIdx.x + off];
            float M  = fmaxf(m1, m2);
            sm[threadIdx.x] = M;
            ss[threadIdx.x] = s1 * __expf(m1 - M) + s2 * __expf(m2 - M);
        }
        __syncthreads();
    }
    if (threadIdx.x == 0) { stats[row * 2] = sm[0]; stats[row * 2 + 1] = ss[0]; }
}

// ---------------------------------------------------------------------------
// Partial context: ctxpart[bh][split][d][e] = sum_{n in split} exp(k[d,n]-max_d) * v[e,n]
// grid: (128 bh, 8 splits); deterministic (no atomics).
// ---------------------------------------------------------------------------
__global__ __launch_bounds__(256) void context_part_kernel(
    const float* __restrict__ kvbuf, const float* __restrict__ stats,
    float* __restrict__ ctxpart)
{
    const int bh = blockIdx.x, split = blockIdx.y;
    const int b = bh >> 2, h = bh & 3;
    const float* kb = kvbuf + ((size_t)b * 256 + h * 32) * NPIX;
    const float* vb = kvbuf + ((size_t)b * 256 + 128 + h * 32) * NPIX;

    __shared__ float smax[32];
    __shared__ float kx[32 * 129];
    __shared__ float vx[32 * 129];

    const int tid = threadIdx.x;
    if (tid < 32) smax[tid] = stats[((b * 128) + h * 32 + tid) * 2];

    const int e = tid & 31, dgrp = tid >> 5;    // thread owns (d=dgrp*4+i, e)
    float acc[4] = {0.f, 0.f, 0.f, 0.f};

    const int n0 = split * 512;
    for (int nb = n0; nb < n0 + 512; nb += 128) {
        __syncthreads();
        for (int i = tid; i < 32 * 128; i += 256) {
            int r = i >> 7, c = i & 127;
            kx[r * 129 + c] = __expf(kb[(size_t)r * NPIX + nb + c] - smax[r]);
            vx[r * 129 + c] = vb[(size_t)r * NPIX + nb + c];
        }
        __syncthreads();
        for (int nn = 0; nn < 128; ++nn) {
            float vv = vx[e * 129 + nn];
            #pragma unroll
            for (int i2 = 0; i2 < 4; ++i2)
                acc[i2] += kx[(dgrp * 4 + i2) * 129 + nn] * vv;
        }
    }
    float* cp = ctxpart + ((size_t)(bh * 8 + split)) * 1024;
    #pragma unroll
    for (int i2 = 0; i2 < 4; ++i2)
        cp[(dgrp * 4 + i2) * 32 + e] = acc[i2];
}

// ---------------------------------------------------------------------------
// Reduce partials -> ctx, fold w_out:  M[b][o][h*32+d] = sum_e w_out[o,h*32+e]*ctx[d,e]/sum_d
// ---------------------------------------------------------------------------
__global__ __launch_bounds__(256) void context_M_kernel(
    const float* __restrict__ ctxpart, const float* __restrict__ stats,
    const float* __restrict__ w_out, float* __restrict__ Mbuf)
{
    const int bh = blockIdx.x, b = bh >> 2, h = bh & 3;
    __shared__ float ctx[32 * 33];
    __shared__ float wsl[128 * 32];
    const int tid = threadIdx.x;

    for (int i = tid; i < 1024; i += 256) {
        int d = i >> 5;
        const float* cp = ctxpart + (size_t)bh * 8 * 1024 + i;
        float s = 0.f;
        #pragma unroll
        for (int p = 0; p < 8; ++p) s += cp[p * 1024];
        float sinv = 1.0f / stats[((b * 128) + h * 32 + d) * 2 + 1];
        ctx[d * 33 + (i & 31)] = s * sinv;
    }
    for (int i = tid; i < 128 * 32; i += 256) {
        int o = i >> 5, ee = i & 31;
        wsl[i] = w_out[o * CDIM + h * 32 + ee];
    }
    __syncthreads();

    const int o = tid >> 1, dbase = (tid & 1) * 16;
    float* Mrow = Mbuf + (size_t)b * CDIM * CDIM + (size_t)o * CDIM + h * 32;
    for (int j = 0; j < 16; ++j) {
        int d = dbase + j;
        float s = 0.f;
        #pragma unroll
        for (int ee = 0; ee < 32; ++ee) s += wsl[o * 32 + ee] * ctx[d * 33 + ee];
        Mrow[d] = s;
    }
}

// ---------------------------------------------------------------------------
// G[b] = M[b] (128x128) @ Wq (128x128)  (Wq = rows 0..127 of w_qkv). One block/batch.
// ---------------------------------------------------------------------------
__global__ __launch_bounds__(256) void compute_G_kernel(
    const float* __restrict__ Mbuf, const float* __restrict__ w_qkv,
    float* __restrict__ Gbuf)
{
    const int b = blockIdx.x, tid = threadIdx.x;
    __shared__ float wq[32 * 129];
    const int o = tid >> 1, cbase = (tid & 1) * 64;
    float acc[64];
    #pragma unroll
    for (int j = 0; j < 64; ++j) acc[j] = 0.f;
    const float* Mrow = Mbuf + (size_t)b * CDIM * CDIM + (size_t)o * CDIM;

    for (int r0 = 0; r0 < 128; r0 += 32) {
        __syncthreads();
        for (int i = tid; i < 32 * 128; i += 256) {
            int r = i >> 7, c = i & 127;
            wq[r * 129 + c] = w_qkv[(size_t)(r0 + r) * CDIM + c];
        }
        __syncthreads();
        for (int rr = 0; rr < 32; ++rr) {
            float mv = Mrow[r0 + rr];
            #pragma unroll
            for (int j = 0; j < 64; ++j) acc[j] += mv * wq[rr * 129 + cbase + j];
        }
    }
    float* Grow = Gbuf + (size_t)b * CDIM * CDIM + (size_t)o * CDIM + cbase;
    for (int j = 0; j < 64; ++j) Grow[j] = acc[j];
}

// ---------------------------------------------------------------------------
extern "C" void kernel_launch(void* const* d_in, const int* in_sizes, int n_in,
                              void* d_out, int out_size, void* d_ws, size_t ws_size,
                              hipStream_t stream)
{
    const float* x      = (const float*)d_in[0];   // 32 x 128 x 64 x 64
    const float* w_qkv  = (const float*)d_in[1];   // 384 x 128
    const float* w_out  = (const float*)d_in[2];   // 128 x 128
    const float* b_out  = (const float*)d_in[3];   // 128
    float* out = (float*)d_out;                    // 32 x 128 x 64 x 64

    float* kv      = (float*)d_ws;                                   // 32*256*4096
    float* stats   = kv + (size_t)32 * 256 * NPIX;                   // 4096*2
    float* ctxpart = stats + 8192;                                   // 128*8*1024
    float* Mbuf    = ctxpart + (size_t)128 * 8 * 1024;               // 32*128*128
    float* Gbuf    = Mbuf + (size_t)32 * CDIM * CDIM;                // 32*128*128
    size_t need = ((size_t)(Gbuf + (size_t)32 * CDIM * CDIM - (float*)d_ws)) * sizeof(float);
    if (ws_size < need) return;  // workspace too small; bail deterministically

    // 1) k,v = w_qkv[128:384] @ x[b]
    wmma_gemm_kernel<<<dim3(64, 4, 32), 256, 0, stream>>>(
        w_qkv + 128 * CDIM, 0L, x, kv, 256L * NPIX, nullptr);
    // 2) softmax stats over k rows
    row_stats_kernel<<<4096, 256, 0, stream>>>(kv, stats);
    // 3) partial context
    context_part_kernel<<<dim3(128, 8), 256, 0, stream>>>(kv, stats, ctxpart);
    // 4) ctx reduce + fold w_out -> M
    context_M_kernel<<<128, 256, 0, stream>>>(ctxpart, stats, w_out, Mbuf);
    // 5) G = M @ Wq
    compute_G_kernel<<<32, 256, 0, stream>>>(Mbuf, w_qkv, Gbuf);
    // 6) out = G @ x + b_out
    wmma_gemm_kernel<<<dim3(64, 2, 32), 256, 0, stream>>>(
        Gbuf, (long)(CDIM * CDIM), x, out, (long)(CDIM * NPIX), b_out);
}